// MultiHiPPOLTIAttention_84842783965437
// MI455X (gfx1250) — compile-verified
//
#include <hip/hip_runtime.h>
#include <math.h>

// MultiHiPPOLTIAttention for MI455X (gfx1250, wave32, WMMA fp32 16x16x4).
//
// Pipeline (all fp32, all GEMMs via V_WMMA_F32_16X16X4_F32):
//   1) P[0] = M = GA^T;  log-depth doubling -> P[j] = M^(j+1), j=0..63
//   2) U[j] = GB @ M^j  (j=0..63)
//   3) local chunk convolutions lc[b,i] = sum_{j<=i} f[bL+j] * U[i-j]   (L=64)
//   4) chunk-level scan: S[b+1] = S[b] @ M^64 + lc[b,63]  (128 serial steps)
//   5) carry application: c[bL+i] = lc[b,i] + S[b] @ M^(i+1)  (64 batched GEMMs)
//   6) scores = q k^T * 0.25 -> written into attn region of d_out
//   7) in-place row softmax (row cached in LDS)
//   8) context = attn @ v (split-K=8, deterministic partials in ws)
//   9) out = context @ Wp + bp (N=1000 edge-masked)

typedef __attribute__((ext_vector_type(2))) float v2f;
typedef __attribute__((ext_vector_type(8))) float v8f;

#define SEQN   8192
#define NS     128
#define CHL    64      // chunk length
#define NCH    128     // number of chunks
#define OUTD   1000

static __device__ __forceinline__ v8f wmma_f32(v2f a, v2f b, v8f c) {
  // (neg_a, A, neg_b, B, c_mod, C, reuse_a, reuse_b)
  return __builtin_amdgcn_wmma_f32_16x16x4_f32(false, a, false, b, (short)0, c,
                                               false, false);
}

// ---------------------------------------------------------------- P0 = GA^T
__global__ void k_transpose(const float* __restrict__ GA, float* __restrict__ P0) {
  int t = blockIdx.x * 256 + threadIdx.x;   // 16384 threads
  int i = t >> 7, j = t & 127;
  P0[i * 128 + j] = GA[j * 128 + i];
}

// Doubling round: for q=1..p:  P[p+q-1] = P[q-1] @ P[p-1]   (M^q * M^p = M^(p+q))
// Grid: p*64 waves (p*8 blocks of 256 threads).
__global__ void k_power_round(float* __restrict__ P, int p) {
  int w    = blockIdx.x * 8 + (threadIdx.x >> 5);
  int lane = threadIdx.x & 31;
  int mm   = w >> 6;                 // q-1
  int tile = w & 63;
  int m0 = (tile >> 3) << 4;
  int n0 = (tile & 7) << 4;
  const float* A = P + mm * 16384;
  const float* B = P + (p - 1) * 16384;
  float*       D = P + (p + mm) * 16384;
  int ml = lane & 15, half = lane >> 4;
  v8f acc = {};
  for (int k0 = 0; k0 < 128; k0 += 4) {
    int ka = k0 + 2 * half;
    v2f a = *(const v2f*)(A + (m0 + ml) * 128 + ka);
    v2f b;
    b.x = B[ka * 128 + n0 + ml];
    b.y = B[(ka + 1) * 128 + n0 + ml];
    acc = wmma_f32(a, b, acc);
  }
#pragma unroll
  for (int r = 0; r < 8; ++r)
    D[(m0 + r + 8 * half) * 128 + n0 + ml] = acc[r];
}

// ------------------------------------------------- U[j] = GB @ M^j, U[0] = GB
__global__ void k_u(const float* __restrict__ GB, const float* __restrict__ P,
                    float* __restrict__ U) {
  int t = blockIdx.x * 256 + threadIdx.x;   // 8192 threads
  int j = t >> 7, n = t & 127;
  float acc;
  if (j == 0) {
    acc = GB[n];
  } else {
    acc = 0.f;
    const float* M = P + (j - 1) * 16384;
    for (int m = 0; m < 128; ++m) acc += GB[m] * M[m * 128 + n];
  }
  U[j * 128 + n] = acc;
}

// ------------------- local per-chunk convolution: c[bL+i] = sum f[j]*U[i-j]
__global__ void k_local_scan(const float* __restrict__ f0, const float* __restrict__ f1,
                             const float* __restrict__ f2, const float* __restrict__ U,
                             float* __restrict__ c0, float* __restrict__ c1,
                             float* __restrict__ c2) {
  int s = blockIdx.x >> 7;        // 0..2 (seq), grid = 384
  int b = blockIdx.x & 127;       // chunk
  const float* f = (s == 0) ? f0 : ((s == 1) ? f1 : f2);
  float*       c = (s == 0) ? c0 : ((s == 1) ? c1 : c2);
  int n = threadIdx.x;            // 128 threads
  __shared__ float fs[CHL];
  if (n < CHL) fs[n] = f[b * CHL + n];
  __syncthreads();
  for (int i = 0; i < CHL; ++i) {
    float acc = 0.f;
    for (int j = 0; j <= i; ++j) acc += fs[j] * U[(i - j) * 128 + n];
    c[(b * CHL + i) * 128 + n] = acc;
  }
}

// ------------- chunk scan (serial, 128 steps): S[b]=state before chunk b
__global__ void k_chunk_scan(const float* __restrict__ c0, const float* __restrict__ c1,
                             const float* __restrict__ c2, const float* __restrict__ P63,
                             float* __restrict__ Sall) {
  int t = threadIdx.x;            // 384 threads
  int s = t >> 7, n = t & 127;
  const float* c = (s == 0) ? c0 : ((s == 1) ? c1 : c2);
  __shared__ float st[3][128];
  st[s][n] = 0.f;
  __syncthreads();
  for (int b = 0; b < NCH; ++b) {
    Sall[(s * 128 + b) * 128 + n] = st[s][n];
    float acc = c[(b * CHL + CHL - 1) * 128 + n];   // local final of chunk b
    for (int m = 0; m < 128; ++m) acc += st[s][m] * P63[m * 128 + n];
    __syncthreads();
    st[s][n] = acc;
    __syncthreads();
  }
}

// ----- carry: for each i: O_i = Sall(384x128) @ P[i](128x128), add into c arrays
// 12288 waves = 64 i * 24 row-tiles * 8 col-tiles
__global__ void k_apply_carry(const float* __restrict__ Sall, const float* __restrict__ P,
                              float* __restrict__ c0, float* __restrict__ c1,
                              float* __restrict__ c2) {
  int w    = blockIdx.x * 8 + (threadIdx.x >> 5);
  int lane = threadIdx.x & 31;
  int i    = w / 192;
  int rest = w % 192;
  int m0 = (rest >> 3) << 4;
  int n0 = (rest & 7) << 4;
  const float* B = P + i * 16384;           // M^(i+1)
  int ml = lane & 15, half = lane >> 4;
  v8f acc = {};
  for (int k0 = 0; k0 < 128; k0 += 4) {
    int ka = k0 + 2 * half;
    v2f a = *(const v2f*)(Sall + (m0 + ml) * 128 + ka);
    v2f b;
    b.x = B[ka * 128 + n0 + ml];
    b.y = B[(ka + 1) * 128 + n0 + ml];
    acc = wmma_f32(a, b, acc);
  }
#pragma unroll
  for (int r = 0; r < 8; ++r) {
    int R   = m0 + r + 8 * half;            // row in Sall: s*128 + chunk
    int s   = R >> 7;
    int bch = R & 127;
    float* c = (s == 0) ? c0 : ((s == 1) ? c1 : c2);
    float* p = c + (bch * CHL + i) * 128 + n0 + ml;
    *p += acc[r];
  }
}

// ----------------- scores = q k^T * 0.25 -> attn region. 32768 waves (512x64).
// Each wave: 16 rows x 128 cols strip, K=128.
__global__ void k_qk(const float* __restrict__ q, const float* __restrict__ kmat,
                     float* __restrict__ S) {
  int w    = blockIdx.x * 8 + (threadIdx.x >> 5);
  int lane = threadIdx.x & 31;
  int m0 = (w & 511) << 4;
  int nb = (w >> 9) << 7;
  int ml = lane & 15, half = lane >> 4;
  v8f acc[8] = {};
  for (int k0 = 0; k0 < 128; k0 += 4) {
    int ka = k0 + 2 * half;
    v2f a = *(const v2f*)(q + (m0 + ml) * 128 + ka);
#pragma unroll
    for (int j = 0; j < 8; ++j) {
      v2f b = *(const v2f*)(kmat + (nb + j * 16 + ml) * 128 + ka);  // NT
      acc[j] = wmma_f32(a, b, acc[j]);
    }
  }
#pragma unroll
  for (int j = 0; j < 8; ++j)
#pragma unroll
    for (int r = 0; r < 8; ++r)
      S[(m0 + r + 8 * half) * SEQN + nb + j * 16 + ml] = acc[j][r] * 0.25f;
}

// ------------------------- in-place row softmax, row cached in LDS (32 KB)
__global__ void k_softmax(float* __restrict__ attn) {
  __shared__ float srow[SEQN];
  __shared__ float red[256];
  float* base = attn + blockIdx.x * SEQN;
  int tid = threadIdx.x;
  float mx = -3.4e38f;
  for (int idx = tid; idx < SEQN; idx += 256) {
    float x = base[idx];
    srow[idx] = x;
    mx = fmaxf(mx, x);
  }
  red[tid] = mx; __syncthreads();
  for (int off = 128; off > 0; off >>= 1) {
    if (tid < off) red[tid] = fmaxf(red[tid], red[tid + off]);
    __syncthreads();
  }
  mx = red[0];
  __syncthreads();
  float sum = 0.f;
  for (int idx = tid; idx < SEQN; idx += 256) {
    float e = __expf(srow[idx] - mx);
    srow[idx] = e;
    sum += e;
  }
  red[tid] = sum; __syncthreads();
  for (int off = 128; off > 0; off >>= 1) {
    if (tid < off) red[tid] += red[tid + off];
    __syncthreads();
  }
  float inv = 1.0f / red[0];
  for (int idx = tid; idx < SEQN; idx += 256) base[idx] = srow[idx] * inv;
}

// -------- context partials = attn @ v, split-K=8. 4096 waves (512 mtiles x 8 ksegs)
__global__ void k_av(const float* __restrict__ attn, const float* __restrict__ v,
                     float* __restrict__ part) {
  int w    = blockIdx.x * 8 + (threadIdx.x >> 5);
  int lane = threadIdx.x & 31;
  int m0 = (w & 511) << 4;
  int kb = (w >> 9) << 10;
  int ml = lane & 15, half = lane >> 4;
  v8f acc[8] = {};
  for (int kk = 0; kk < 1024; kk += 4) {
    int ka = kb + kk + 2 * half;
    v2f a = *(const v2f*)(attn + (m0 + ml) * SEQN + ka);
#pragma unroll
    for (int j = 0; j < 8; ++j) {
      v2f b;                                   // NN: column of v
      b.x = v[ka * 128 + j * 16 + ml];
      b.y = v[(ka + 1) * 128 + j * 16 + ml];
      acc[j] = wmma_f32(a, b, acc[j]);
    }
  }
  float* dst = part + (w >> 9) * 1048576;
#pragma unroll
  for (int j = 0; j < 8; ++j)
#pragma unroll
    for (int r = 0; r < 8; ++r)
      dst[(m0 + r + 8 * half) * 128 + j * 16 + ml] = acc[j][r];
}

__global__ void k_reduce(const float* __restrict__ part, float* __restrict__ ctx) {
  int t = blockIdx.x * 256 + threadIdx.x;     // 1048576 threads
  float s = 0.f;
#pragma unroll
  for (int g = 0; g < 8; ++g) s += part[g * 1048576 + t];
  ctx[t] = s;
}

// ------------- out = ctx @ Wp + bp. 32256 waves = 512 mtiles * 63 ntiles (N=1000)
__global__ void k_proj(const float* __restrict__ ctx, const float* __restrict__ Wp,
                       const float* __restrict__ bp, float* __restrict__ out) {
  int w    = blockIdx.x * 8 + (threadIdx.x >> 5);
  int lane = threadIdx.x & 31;
  int mt = w / 63;
  int nt = w % 63;
  int m0 = mt << 4;
  int ml = lane & 15, half = lane >> 4;
  int n  = nt * 16 + ml;
  bool ok = (n < OUTD);
  v8f acc = {};
  for (int k0 = 0; k0 < 128; k0 += 4) {
    int ka = k0 + 2 * half;
    v2f a = *(const v2f*)(ctx + (m0 + ml) * 128 + ka);
    v2f b;
    b.x = ok ? Wp[ka * OUTD + n] : 0.f;
    b.y = ok ? Wp[(ka + 1) * OUTD + n] : 0.f;
    acc = wmma_f32(a, b, acc);
  }
  if (ok) {
    float bias = bp[n];
#pragma unroll
    for (int r = 0; r < 8; ++r)
      out[(m0 + r + 8 * half) * OUTD + n] = acc[r] + bias;
  }
}

extern "C" void kernel_launch(void* const* d_in, const int* in_sizes, int n_in,
                              void* d_out, int out_size, void* d_ws, size_t ws_size,
                              hipStream_t stream) {
  const float* fq = (const float*)d_in[0];
  const float* fk = (const float*)d_in[1];
  const float* fv = (const float*)d_in[2];
  const float* GA = (const float*)d_in[3];
  const float* GB = (const float*)d_in[4];
  const float* Wp = (const float*)d_in[5];
  const float* bp = (const float*)d_in[6];

  float* out  = (float*)d_out;
  float* attn = out + SEQN * OUTD;            // outputs: (out, attn) concatenated

  float* ws   = (float*)d_ws;
  float* P    = ws;                           // 64 * 16384        (M^1..M^64)
  float* U    = P + 64 * 16384;               // 64 * 128
  float* cq   = U + 64 * 128;                 // 8192 * 128
  float* ck   = cq + SEQN * NS;
  float* cv   = ck + SEQN * NS;
  float* Sall = cv + SEQN * NS;               // 384 * 128
  float* part = Sall + 384 * NS;              // 8 * 8192 * 128
  float* ctx  = part + 8 * SEQN * NS;         // 8192 * 128

  k_transpose<<<64, 256, 0, stream>>>(GA, P);
  for (int p = 1; p < 64; p <<= 1)
    k_power_round<<<p * 8, 256, 0, stream>>>(P, p);
  k_u<<<32, 256, 0, stream>>>(GB, P, U);
  k_local_scan<<<384, 128, 0, stream>>>(fq, fk, fv, U, cq, ck, cv);
  k_chunk_scan<<<1, 384, 0, stream>>>(cq, ck, cv, P + 63 * 16384, Sall);
  k_apply_carry<<<1536, 256, 0, stream>>>(Sall, P, cq, ck, cv);
  k_qk<<<4096, 256, 0, stream>>>(cq, ck, attn);
  k_softmax<<<SEQN, 256, 0, stream>>>(attn);
  k_av<<<512, 256, 0, stream>>>(attn, cv, part);
  k_reduce<<<4096, 256, 0, stream>>>(part, ctx);
  k_proj<<<4032, 256, 0, stream>>>(ctx, Wp, bp, out);
}